// EnhancedQuantumPINN_62371515072641
// MI455X (gfx1250) — compile-verified
//
#include <hip/hip_runtime.h>
#include <hip/hip_bf16.h>
#include <math.h>

#define NQ   4
#define DIM  16
#define NL   5
#define BLK  256

typedef __attribute__((ext_vector_type(2))) float v2f;
typedef __attribute__((ext_vector_type(8))) float v8f;

// CDNA5 has a hardware tanh transcendental (V_TANH_F32). Prefer the clang
// builtin when declared; otherwise emit it directly. The raw-asm path adds a
// v_nop to respect the TRANS-op result/source hazard rule (ISA 7.4), since
// inline asm bypasses compiler hazard insertion.
__device__ __forceinline__ float fast_tanh(float x) {
#if __has_builtin(__builtin_amdgcn_tanhf)
    return __builtin_amdgcn_tanhf(x);
#elif __has_builtin(__builtin_amdgcn_tanh_f32)
    return __builtin_amdgcn_tanh_f32(x);
#else
    float r;
    asm volatile("v_tanh_f32 %0, %1\n\tv_nop" : "=v"(r) : "v"(x));
    return r;
#endif
}

__global__ __launch_bounds__(BLK) void qpinn_kernel(
    const float* __restrict__ xy, const float* __restrict__ W1, const float* __restrict__ b1,
    const float* __restrict__ W2, const float* __restrict__ b2, const float* __restrict__ W3,
    const float* __restrict__ b3, const float* __restrict__ W4, const float* __restrict__ b4,
    float* __restrict__ out, int B)
{
    __shared__ float hS[BLK][17];    // h activations, padded (17 coprime to 64 banks)
    __shared__ float w2S[16][40];    // W2 staged once per block
    __shared__ float pS[BLK][41];    // tanh(h@W2+b2), padded stride 41

    const int tid   = threadIdx.x;
    const int lane  = tid & 31;
    const int wv    = tid >> 5;
    const int half  = lane >> 4;     // 0: lanes 0-15, 1: lanes 16-31
    const int l16   = lane & 15;
    const int bIdx  = blockIdx.x * BLK + tid;
    const int bSafe = (bIdx < B) ? bIdx : (B - 1);

    // ---- stage W2 into LDS ----
    for (int i = tid; i < 16 * 40; i += BLK) w2S[i / 40][i % 40] = W2[i];

    // ---- stage 1: h = tanh(xy @ W1 + b1), per-thread, into LDS ----
    const float x = xy[2 * bSafe + 0];
    const float y = xy[2 * bSafe + 1];
    #pragma unroll
    for (int j = 0; j < 16; ++j)
        hS[tid][j] = fast_tanh(fmaf(x, W1[j], fmaf(y, W1[16 + j], b1[j])));

    __syncthreads();

    // ---- stage 2: params = tanh(h @ W2 + b2) via V_WMMA_F32_16X16X4_F32 ----
    // Each wave computes its own 32 batch rows x 40 cols.
    const int waveBase = wv * 32;
    #pragma unroll
    for (int m = 0; m < 2; ++m) {
        // A fragment rows: lanes 0-15 and 16-31 both address M = l16 (ISA 7.12.2)
        const int arow = waveBase + m * 16 + l16;
        #pragma unroll
        for (int n = 0; n < 3; ++n) {
            const int col    = n * 16 + l16;     // output N index for this lane
            const bool colOK = (col < 40);
            const float bias = colOK ? b2[col] : 0.0f;
            v8f acc;
            #pragma unroll
            for (int r = 0; r < 8; ++r) acc[r] = bias;  // C init: bias depends only on N

            #pragma unroll
            for (int kc = 0; kc < 4; ++kc) {
                // f32 A 16x4: lanes 0-15 hold K=k0,k0+1; lanes 16-31 hold K=k0+2,k0+3
                const int k0 = kc * 4 + half * 2;
                v2f a, bm;
                a[0]  = hS[arow][k0];
                a[1]  = hS[arow][k0 + 1];
                bm[0] = colOK ? w2S[k0][col]     : 0.0f;
                bm[1] = colOK ? w2S[k0 + 1][col] : 0.0f;
                acc = __builtin_amdgcn_wmma_f32_16x16x4_f32(
                          false, a, false, bm, (short)0, acc, false, false);
            }
            // D layout: VGPR r -> M = half*8 + r, N = l16
            if (colOK) {
                #pragma unroll
                for (int r = 0; r < 8; ++r)
                    pS[waveBase + m * 16 + half * 8 + r][col] = fast_tanh(acc[r]);
            }
        }
    }
    __syncthreads();

    // ---- stage 3: 4-qubit state-vector sim, fully register resident ----
    float re[DIM], im[DIM];
    {
        // H,RY(x*pi),RZ(y*pi) on every qubit from |0>: identical product-state factor.
        const float PIv = 3.14159265358979323846f;
        const float sn = __sinf(x * PIv * 0.5f);
        const float cs = __cosf(x * PIv * 0.5f);
        const float is2 = 0.70710678118654752440f;
        const float a0 = is2 * (cs - sn), a1 = is2 * (cs + sn);
        const float su = __sinf(y * PIv * 0.5f);
        const float cu = __cosf(y * PIv * 0.5f);
        const float A0r = a0 * cu, A0i = -a0 * su;   // a0 * e^{-iu}
        const float A1r = a1 * cu, A1i =  a1 * su;   // a1 * e^{+iu}
        float p0r[5], p0i[5], p1r[5], p1i[5];
        p0r[0] = 1.f; p0i[0] = 0.f; p1r[0] = 1.f; p1i[0] = 0.f;
        #pragma unroll
        for (int k = 1; k <= 4; ++k) {
            p0r[k] = p0r[k-1]*A0r - p0i[k-1]*A0i;
            p0i[k] = p0r[k-1]*A0i + p0i[k-1]*A0r;
            p1r[k] = p1r[k-1]*A1r - p1i[k-1]*A1i;
            p1i[k] = p1r[k-1]*A1i + p1i[k-1]*A1r;
        }
        #pragma unroll
        for (int d = 0; d < DIM; ++d) {
            const int p = __popc(d);
            re[d] = p0r[4-p]*p1r[p] - p0i[4-p]*p1i[p];
            im[d] = p0r[4-p]*p1i[p] + p0i[4-p]*p1r[p];
        }
    }

    for (int l = 0; l < NL; ++l) {
        #pragma unroll
        for (int qb = 0; qb < 4; ++qb) {
            const float ax = pS[tid][l * 8 + qb];       // RX angle
            const float ay = pS[tid][l * 8 + qb + 4];   // RY angle
            const float sa = __sinf(ax * 0.5f), ca = __cosf(ax * 0.5f);
            const float sb = __sinf(ay * 0.5f), cb = __cosf(ay * 0.5f);
            // U = RY(ay)*RX(ax): U00=(A,Bv) U01=(-Cv,-Dv) U10=(Cv,-Dv) U11=(A,-Bv)
            const float A  = cb * ca;
            const float Bv = sb * sa;
            const float Cv = sb * ca;
            const float Dv = cb * sa;
            const int  R   = 8 >> qb;                   // qubit qb <-> index bit (3-qb)
            #pragma unroll
            for (int i0 = 0; i0 < DIM; ++i0) {
                if (i0 & R) continue;
                const int i1 = i0 | R;
                const float r0 = re[i0], m0 = im[i0];
                const float r1 = re[i1], m1 = im[i1];
                re[i0] = A*r0 - Bv*m0 - Cv*r1 + Dv*m1;
                im[i0] = A*m0 + Bv*r0 - Cv*m1 - Dv*r1;
                re[i1] = Cv*r0 + Dv*m0 + A*r1 + Bv*m1;
                im[i1] = Cv*m0 - Dv*r0 + A*m1 - Bv*r1;
            }
        }
        // CZ ring (0,1)(1,2)(2,3)(3,0): net sign -1 exactly for d in {3,6,9,12}
        re[3]  = -re[3];  im[3]  = -im[3];
        re[6]  = -re[6];  im[6]  = -im[6];
        re[9]  = -re[9];  im[9]  = -im[9];
        re[12] = -re[12]; im[12] = -im[12];
    }

    // ---- stage 4: q = probs @ Z^T, then tanh(q@W3+b3)@W4+b4 ----
    float q0 = 0.f, q1 = 0.f, q2 = 0.f, q3 = 0.f;
    #pragma unroll
    for (int d = 0; d < DIM; ++d) {
        const float pr = re[d]*re[d] + im[d]*im[d];
        q0 += (d & 8) ? -pr : pr;   // qubit 0 <-> bit 3
        q1 += (d & 4) ? -pr : pr;
        q2 += (d & 2) ? -pr : pr;
        q3 += (d & 1) ? -pr : pr;
    }
    float o = b4[0];
    #pragma unroll
    for (int j = 0; j < 8; ++j) {
        const float acc = b3[j] + q0*W3[j] + q1*W3[8 + j] + q2*W3[16 + j] + q3*W3[24 + j];
        o += fast_tanh(acc) * W4[j];
    }
    if (bIdx < B) out[bIdx] = o;
}

extern "C" void kernel_launch(void* const* d_in, const int* in_sizes, int n_in,
                              void* d_out, int out_size, void* d_ws, size_t ws_size,
                              hipStream_t stream) {
    const float* xy = (const float*)d_in[0];
    const float* W1 = (const float*)d_in[1];
    const float* b1 = (const float*)d_in[2];
    const float* W2 = (const float*)d_in[3];
    const float* b2 = (const float*)d_in[4];
    const float* W3 = (const float*)d_in[5];
    const float* b3 = (const float*)d_in[6];
    const float* W4 = (const float*)d_in[7];
    const float* b4 = (const float*)d_in[8];
    float* out = (float*)d_out;

    const int B = in_sizes[0] / 2;
    const int blocks = (B + BLK - 1) / BLK;
    qpinn_kernel<<<blocks, BLK, 0, stream>>>(xy, W1, b1, W2, b2, W3, b3, W4, b4, out, B);
}